// GATEmbed_85048942396228
// MI455X (gfx1250) — compile-verified
//
#include <hip/hip_runtime.h>
#include <math.h>

typedef __attribute__((ext_vector_type(2))) float v2f;
typedef __attribute__((ext_vector_type(8))) float v8f;

#define NEG_SLOPE 0.2f

// ---------------------------------------------------------------------------
// fp32 GEMM via V_WMMA_F32_16X16X4_F32.  One wave computes one 16x16 C tile,
// looping K in steps of 4.  M (=50000) is a multiple of 16 so EXEC is all-1s
// at every WMMA (ISA requirement).
// A: [M,K] row-major, W: [K,NC] row-major, C: [M,NC] row-major.
// ---------------------------------------------------------------------------
__global__ void wmma_gemm_f32(const float* __restrict__ A,
                              const float* __restrict__ W,
                              float* __restrict__ C,
                              int M, int K, int NC) {
    const int wave = (blockIdx.x * blockDim.x + threadIdx.x) >> 5;
    const int lane = threadIdx.x & 31;
    const int ctiles = NC >> 4;
    const int rt = wave / ctiles;
    const int ct = wave % ctiles;
    const int row0 = rt << 4;
    if (row0 >= M) return;                 // wave-uniform exit

    const int half = lane >> 4;            // 0: lanes 0-15, 1: lanes 16-31
    const int l15  = lane & 15;
    const int arow = row0 + l15;           // always < M (M % 16 == 0)
    const int col  = (ct << 4) + l15;

    v8f acc = {};
    for (int k = 0; k < K; k += 4) {
        const int kk = k + 2 * half;
        // A frag: 16x4, lane l15 holds row, VGPR0=K=kk, VGPR1=K=kk+1
        v2f a;
        a.x = A[(size_t)arow * K + kk];
        a.y = A[(size_t)arow * K + kk + 1];
        // B frag: 4x16, mirrored K split across lane halves
        v2f b;
        b.x = W[(size_t)kk * NC + col];
        b.y = W[(size_t)(kk + 1) * NC + col];
        acc = __builtin_amdgcn_wmma_f32_16x16x4_f32(
            /*neg_a=*/false, a, /*neg_b=*/false, b,
            /*c_mod=*/(short)0, acc, /*reuse_a=*/false, /*reuse_b=*/false);
    }
    // C/D layout: VGPR v, lane -> row = row0 + v + 8*half, col = col
#pragma unroll
    for (int v = 0; v < 8; ++v) {
        const int r = row0 + v + 8 * half;
        C[(size_t)r * NC + col] = acc[v];
    }
}

// ---------------------------------------------------------------------------
// Zero-fill scratch accumulator (d_ws is poisoned; must re-init every call).
// ---------------------------------------------------------------------------
__global__ void fill_zero(float* __restrict__ p, long long n) {
    long long i = (long long)blockIdx.x * blockDim.x + threadIdx.x;
    if (i < n) p[i] = 0.0f;
}

// ---------------------------------------------------------------------------
// Per-(node,head) attention logits + init of segment max / sum buffers.
// ---------------------------------------------------------------------------
__global__ void att_logits(const float* __restrict__ h,
                           const float* __restrict__ a_src,
                           const float* __restrict__ a_dst,
                           float* __restrict__ al_src,
                           float* __restrict__ al_dst,
                           float* __restrict__ m,
                           float* __restrict__ s,
                           int N, int H, int CH) {
    int i = blockIdx.x * blockDim.x + threadIdx.x;
    if (i >= N * H) return;
    const int n = i / H, hd = i % H;
    const float* hp = h + (size_t)n * H * CH + (size_t)hd * CH;
    const float* as = a_src + hd * CH;
    const float* ad = a_dst + hd * CH;
    float s0 = 0.f, s1 = 0.f;
    for (int c = 0; c < CH; ++c) {
        const float v = hp[c];
        s0 += v * as[c];
        s1 += v * ad[c];
    }
    al_src[i] = s0;
    al_dst[i] = s1;
    m[i] = -__builtin_inff();
    s[i] = 0.0f;
}

// Float atomic max via signed-max / unsigned-min bit trick (valid with -inf init,
// correct under mixed sign interleavings).
__device__ __forceinline__ void atomicMaxFloat(float* addr, float val) {
    if (val >= 0.0f)
        atomicMax((int*)addr, __float_as_int(val));
    else
        atomicMin((unsigned int*)addr, __float_as_uint(val));
}

__device__ __forceinline__ float lrelu(float x) {
    return x > 0.0f ? x : NEG_SLOPE * x;
}

// ---------------------------------------------------------------------------
// Pass 1 over edges (incl. implicit self loops for e >= E): segment max of
// leaky_relu(al_src[src] + al_dst[dst]) into m[dst].
// ---------------------------------------------------------------------------
__global__ void edge_max(const long long* __restrict__ ei, int E, int N, int H,
                         const float* __restrict__ al_src,
                         const float* __restrict__ al_dst,
                         float* __restrict__ m) {
    int e = blockIdx.x * blockDim.x + threadIdx.x;
    if (e >= E + N) return;
    int src, dst;
    if (e < E) { src = (int)ei[e]; dst = (int)ei[(size_t)E + e]; }
    else       { src = dst = e - E; }
    for (int hd = 0; hd < H; ++hd) {
        const float x = lrelu(al_src[src * H + hd] + al_dst[dst * H + hd]);
        atomicMaxFloat(&m[dst * H + hd], x);
    }
}

// ---------------------------------------------------------------------------
// Pass 2: segment sum of exp(e - m[dst]) into s[dst].
// ---------------------------------------------------------------------------
__global__ void edge_sum(const long long* __restrict__ ei, int E, int N, int H,
                         const float* __restrict__ al_src,
                         const float* __restrict__ al_dst,
                         const float* __restrict__ m,
                         float* __restrict__ s) {
    int e = blockIdx.x * blockDim.x + threadIdx.x;
    if (e >= E + N) return;
    int src, dst;
    if (e < E) { src = (int)ei[e]; dst = (int)ei[(size_t)E + e]; }
    else       { src = dst = e - E; }
    for (int hd = 0; hd < H; ++hd) {
        const float x  = lrelu(al_src[src * H + hd] + al_dst[dst * H + hd]);
        const float ex = __expf(x - m[dst * H + hd]);
        atomicAdd(&s[dst * H + hd], ex);
    }
}

// ---------------------------------------------------------------------------
// Pass 3: weighted gather/scatter.  Each thread handles one edge x 4 channels
// of the flattened [H*CH] feature (recomputes alpha from L2-resident stats).
// ---------------------------------------------------------------------------
__global__ void edge_aggregate(const long long* __restrict__ ei, int E, int N,
                               int H, int CH,
                               const float* __restrict__ al_src,
                               const float* __restrict__ al_dst,
                               const float* __restrict__ m,
                               const float* __restrict__ s,
                               const float* __restrict__ hfeat,
                               float* __restrict__ agg) {
    const int HC  = H * CH;
    const int tpe = HC >> 2;               // threads per edge (4 channels each)
    long long gid = (long long)blockIdx.x * blockDim.x + threadIdx.x;
    long long e   = gid / tpe;
    const int part = (int)(gid % tpe);
    if (e >= (long long)E + N) return;
    int src, dst;
    if (e < E) { src = (int)ei[e]; dst = (int)ei[(size_t)E + e]; }
    else       { src = dst = (int)(e - E); }
    const int c0 = part << 2;
    const int hd = c0 / CH;
    const float x     = lrelu(al_src[src * H + hd] + al_dst[dst * H + hd]);
    const float alpha = __expf(x - m[dst * H + hd]) / (s[dst * H + hd] + 1e-16f);
    const float* hs = hfeat + (size_t)src * HC + c0;
    float* ap       = agg   + (size_t)dst * HC + c0;
#pragma unroll
    for (int j = 0; j < 4; ++j)
        atomicAdd(&ap[j], hs[j] * alpha);
}

// out = elu(out + bias), in place (becomes layer-2 input).
__global__ void bias_elu(float* __restrict__ a, const float* __restrict__ b,
                         long long total, int HC) {
    long long i = (long long)blockIdx.x * blockDim.x + threadIdx.x;
    if (i >= total) return;
    const float v = a[i] + b[(int)(i % HC)];
    a[i] = v > 0.0f ? v : expm1f(v);
}

// d_out = agg2 + bias2
__global__ void bias_out(const float* __restrict__ a, const float* __restrict__ b,
                         float* __restrict__ out, long long total, int HC) {
    long long i = (long long)blockIdx.x * blockDim.x + threadIdx.x;
    if (i >= total) return;
    out[i] = a[i] + b[(int)(i % HC)];
}

// ---------------------------------------------------------------------------
extern "C" void kernel_launch(void* const* d_in, const int* in_sizes, int n_in,
                              void* d_out, int out_size, void* d_ws, size_t ws_size,
                              hipStream_t stream) {
    const float*     x      = (const float*)d_in[0];
    const long long* ei     = (const long long*)d_in[1];
    const float*     W1     = (const float*)d_in[2];
    const float*     a_src1 = (const float*)d_in[3];
    const float*     a_dst1 = (const float*)d_in[4];
    const float*     b1     = (const float*)d_in[5];
    const float*     W2     = (const float*)d_in[6];
    const float*     a_src2 = (const float*)d_in[7];
    const float*     a_dst2 = (const float*)d_in[8];
    const float*     b2     = (const float*)d_in[9];

    const int N = in_sizes[0] / 128;   // 50000
    const int E = in_sizes[1] / 2;     // 1600000
    const int ET = E + N;              // edges incl. self loops

    // workspace layout (floats)
    float* ws    = (float*)d_ws;
    float* h1    = ws;                           // N*128
    float* agg1  = h1   + (size_t)N * 128;       // N*128 (becomes elu'd layer-2 input)
    float* h2    = agg1 + (size_t)N * 128;       // N*64
    float* agg2  = h2   + (size_t)N * 64;        // N*64
    float* al_s1 = agg2 + (size_t)N * 64;        // N*4
    float* al_d1 = al_s1 + (size_t)N * 4;
    float* m1    = al_d1 + (size_t)N * 4;
    float* s1    = m1    + (size_t)N * 4;
    float* al_s2 = s1    + (size_t)N * 4;        // N
    float* al_d2 = al_s2 + N;
    float* m2    = al_d2 + N;
    float* s2    = m2    + N;

    const int TB = 256;
    auto blocks1d = [](long long t, int tb) { return (int)((t + tb - 1) / tb); };

    // ---------------- Layer 1: GATConv(128 -> 32 x 4 heads) ----------------
    {
        const int waves = ((N + 15) / 16) * (128 / 16);
        wmma_gemm_f32<<<blocks1d((long long)waves * 32, TB), TB, 0, stream>>>(
            x, W1, h1, N, 128, 128);
    }
    fill_zero<<<blocks1d((long long)N * 128, TB), TB, 0, stream>>>(agg1, (long long)N * 128);
    att_logits<<<blocks1d((long long)N * 4, TB), TB, 0, stream>>>(
        h1, a_src1, a_dst1, al_s1, al_d1, m1, s1, N, 4, 32);
    edge_max<<<blocks1d(ET, TB), TB, 0, stream>>>(ei, E, N, 4, al_s1, al_d1, m1);
    edge_sum<<<blocks1d(ET, TB), TB, 0, stream>>>(ei, E, N, 4, al_s1, al_d1, m1, s1);
    edge_aggregate<<<blocks1d((long long)ET * 32, TB), TB, 0, stream>>>(
        ei, E, N, 4, 32, al_s1, al_d1, m1, s1, h1, agg1);
    bias_elu<<<blocks1d((long long)N * 128, TB), TB, 0, stream>>>(
        agg1, b1, (long long)N * 128, 128);

    // ---------------- Layer 2: GATConv(128 -> 64, 1 head) ------------------
    {
        const int waves = ((N + 15) / 16) * (64 / 16);
        wmma_gemm_f32<<<blocks1d((long long)waves * 32, TB), TB, 0, stream>>>(
            agg1, W2, h2, N, 128, 64);
    }
    fill_zero<<<blocks1d((long long)N * 64, TB), TB, 0, stream>>>(agg2, (long long)N * 64);
    att_logits<<<blocks1d(N, TB), TB, 0, stream>>>(
        h2, a_src2, a_dst2, al_s2, al_d2, m2, s2, N, 1, 64);
    edge_max<<<blocks1d(ET, TB), TB, 0, stream>>>(ei, E, N, 1, al_s2, al_d2, m2);
    edge_sum<<<blocks1d(ET, TB), TB, 0, stream>>>(ei, E, N, 1, al_s2, al_d2, m2, s2);
    edge_aggregate<<<blocks1d((long long)ET * 16, TB), TB, 0, stream>>>(
        ei, E, N, 1, 64, al_s2, al_d2, m2, s2, h2, agg2);
    bias_out<<<blocks1d((long long)N * 64, TB), TB, 0, stream>>>(
        agg2, b2, (float*)d_out, (long long)N * 64, 64);
}